// ImageTextMatching_24412594110965
// MI455X (gfx1250) — compile-verified
//
#include <hip/hip_runtime.h>
#include <stdint.h>

// Problem constants (from reference)
#define NROWS 16384
#define DDIM  2048
#define PDIM  256
#define NB    32
#define NL    100
#define NSEG  (NB * NL)       // 3200 (batch,label) segments
#define SROWS (NB + NSEG)     // 3232 rows in stacked segment-sum matrix S
#define SROWS_P 3328          // padded to 26 * 128 for the block GEMM
#define KB    32              // K-block staged in LDS
#define BSTR  36              // padded LDS stride in floats (144B: 16B-aligned, bank-conflict-free)

typedef __attribute__((ext_vector_type(2))) float v2f;
typedef __attribute__((ext_vector_type(8))) float v8f;

#if __has_builtin(__builtin_amdgcn_global_load_async_to_lds_b128) && \
    __has_builtin(__builtin_amdgcn_s_wait_asynccnt)
#define USE_ASYNC 1
#else
#define USE_ASYNC 0
#endif

#if USE_ASYNC
// Builtin signature (from hipcc diagnostic): param0 is
// 'int __attribute__((vector_size(16))) __device__ *'  == int4 addrspace(1)*.
typedef int v4i_gcc __attribute__((vector_size(16)));
typedef __attribute__((address_space(1))) v4i_gcc gv4i_t;
typedef __attribute__((address_space(3))) v4i_gcc lv4i_t;
__device__ __forceinline__ void async_copy_b128(const float* g, float* l) {
  // generic LDS pointer: low 32 bits are the LDS byte offset (aperture rule)
  __builtin_amdgcn_global_load_async_to_lds_b128(
      (gv4i_t*)(uintptr_t)g, (lv4i_t*)(unsigned)(uintptr_t)l, 0, 0);
}
#endif

__device__ __forceinline__ v8f wmma4(v2f a, v2f b, v8f c) {
  // D = A(16x4, f32) x B(4x16, f32) + C(16x16, f32)
  return __builtin_amdgcn_wmma_f32_16x16x4_f32(false, a, false, b, (short)0, c,
                                               false, false);
}

// ---------------------------------------------------------------------------
// 1) Per-segment counts: cnt_bl[3200], cnt_b[32]
// ---------------------------------------------------------------------------
__global__ void seg_count_kernel(const int* __restrict__ label,
                                 const int* __restrict__ lbatch,
                                 unsigned* __restrict__ cntB,
                                 unsigned* __restrict__ cntBL) {
  int i = blockIdx.x * blockDim.x + threadIdx.x;  // < NROWS
  int b = lbatch[i];
  int s = b * NL + label[i];
  atomicAdd(&cntBL[s], 1u);
  atomicAdd(&cntB[b], 1u);
}

// ---------------------------------------------------------------------------
// 2) Segment sums of x into S rows [32, 3232): one block per input row.
// ---------------------------------------------------------------------------
__global__ void seg_scatter_kernel(const float* __restrict__ x,
                                   const int* __restrict__ label,
                                   const int* __restrict__ lbatch,
                                   float* __restrict__ S) {
  int row = blockIdx.x;
  int b = lbatch[row];
  int s = b * NL + label[row];
  const float* __restrict__ src = x + (size_t)row * DDIM;
  float* __restrict__ dst = S + (size_t)(NB + s) * DDIM;
  int c = threadIdx.x * 4;
#pragma unroll
  for (int h = 0; h < 2; ++h) {
    int col = c + h * 1024;
    float4 v = *(const float4*)(src + col);
    unsafeAtomicAdd(dst + col + 0, v.x);
    unsafeAtomicAdd(dst + col + 1, v.y);
    unsafeAtomicAdd(dst + col + 2, v.z);
    unsafeAtomicAdd(dst + col + 3, v.w);
  }
}

// ---------------------------------------------------------------------------
// 3) Per-batch sums: S[b] = sum over 100 labels of S[32 + b*100 + l]
// ---------------------------------------------------------------------------
__global__ void batch_reduce_kernel(float* __restrict__ S) {
  int idx = blockIdx.x * blockDim.x + threadIdx.x;  // < NB*DDIM
  int b = idx >> 11;
  int col = idx & (DDIM - 1);
  const float* __restrict__ base = S + ((size_t)NB + (size_t)b * NL) * DDIM + col;
  float acc = 0.f;
#pragma unroll 4
  for (int l = 0; l < NL; ++l) acc += base[(size_t)l * DDIM];
  S[(size_t)b * DDIM + col] = acc;
}

// ---------------------------------------------------------------------------
// Stage a 64-col x 32-K weight tile into LDS (column-major, stride BSTR).
// 256 threads: each copies 8 consecutive K-floats of one column (2 x b128).
// ---------------------------------------------------------------------------
__device__ __forceinline__ void stage_tile(const float* __restrict__ Wmat,
                                           int pBase, int kb, float* buf) {
  const int tid = threadIdx.x;
  const int col = tid >> 2;          // 0..63
  const int k0 = (tid & 3) * 8;      // 0,8,16,24
  const float* g = Wmat + (size_t)(pBase + col) * DDIM + kb + k0;
  float* l = buf + col * BSTR + k0;  // byte offset col*144 + k0*4 : 16B aligned
#if USE_ASYNC
  async_copy_b128(g, l);
  async_copy_b128(g + 4, l + 4);
#else
  float4 a = *(const float4*)g;
  float4 b = *(const float4*)(g + 4);
  *(float4*)l = a;
  *(float4*)(l + 4) = b;
#endif
}

// ---------------------------------------------------------------------------
// Block GEMM core: block = 8 waves = 128 rows x 64 cols.
// Wave w computes rows [mBlock + 16w, +16) x cols [pBase, +64) as 4 WMMA tiles.
// B comes from a double-buffered LDS tile; A streams from global (L2-resident).
// ---------------------------------------------------------------------------
__device__ __forceinline__ void block_gemm(const float* __restrict__ Abase,
                                           const float* __restrict__ Wmat,
                                           int mBlock, int pBase,
                                           float (&Bs)[2][64 * BSTR],
                                           v8f acc[4]) {
  const int lane = threadIdx.x & 31;
  const int wave = threadIdx.x >> 5;
  const int kOff = (lane >> 4) * 2;  // lanes 16-31 hold K+2,K+3
  const int nl = lane & 15;
  const float* Arow = Abase + (size_t)(mBlock + wave * 16 + nl) * DDIM + kOff;

  stage_tile(Wmat, pBase, 0, Bs[0]);
  for (int kb = 0; kb < DDIM; kb += KB) {
    const int cur = (kb / KB) & 1;
    if (kb + KB < DDIM) {
      stage_tile(Wmat, pBase, kb + KB, Bs[cur ^ 1]);
#if USE_ASYNC
      __builtin_amdgcn_s_wait_asynccnt(2);  // current buffer's copies done
#endif
    } else {
#if USE_ASYNC
      __builtin_amdgcn_s_wait_asynccnt(0);
#endif
    }
    __syncthreads();
    const float* bs = Bs[cur];
#pragma unroll
    for (int k = 0; k < KB; k += 4) {
      v2f a = *(const v2f*)(Arow + kb + k);
      const int kk = k + kOff;
      v2f b0 = *(const v2f*)(bs + (nl + 0) * BSTR + kk);
      v2f b1 = *(const v2f*)(bs + (nl + 16) * BSTR + kk);
      v2f b2 = *(const v2f*)(bs + (nl + 32) * BSTR + kk);
      v2f b3 = *(const v2f*)(bs + (nl + 48) * BSTR + kk);
      acc[0] = wmma4(a, b0, acc[0]);
      acc[1] = wmma4(a, b1, acc[1]);
      acc[2] = wmma4(a, b2, acc[2]);
      acc[3] = wmma4(a, b3, acc[3]);
    }
    __syncthreads();  // all reads of 'cur' done before it is restaged
  }
}

// ---------------------------------------------------------------------------
// 4) Y = S @ W2^T   (3328p x 2048) x (2048 x 256) -> (3328p x 256)
// ---------------------------------------------------------------------------
__global__ void gemm_seg_kernel(const float* __restrict__ S,
                                const float* __restrict__ W2w,
                                float* __restrict__ Y) {
  __shared__ float Bs[2][64 * BSTR];
  const int mBlock = blockIdx.x * 128;
  const int pBase = blockIdx.y * 64;
  v8f acc[4] = {};
  block_gemm(S, W2w, mBlock, pBase, Bs, acc);

  const int lane = threadIdx.x & 31;
  const int wave = threadIdx.x >> 5;
  const int rhi = (lane >> 4) * 8;  // lanes 16-31 hold M = r+8
  const int n = lane & 15;
#pragma unroll
  for (int t = 0; t < 4; ++t)
#pragma unroll
    for (int r = 0; r < 8; ++r)
      Y[(size_t)(mBlock + wave * 16 + r + rhi) * PDIM + pBase + t * 16 + n] =
          acc[t][r];
}

// ---------------------------------------------------------------------------
// 5) out = x @ W1^T + b1 + b2 + scale * (Y[batch] - Y[32+seg]),  fused epilogue
// ---------------------------------------------------------------------------
__global__ void gemm_main_kernel(const float* __restrict__ x,
                                 const float* __restrict__ W1w,
                                 const float* __restrict__ W1b,
                                 const float* __restrict__ W2b,
                                 const float* __restrict__ Y,
                                 const unsigned* __restrict__ cntB,
                                 const unsigned* __restrict__ cntBL,
                                 const int* __restrict__ label,
                                 const int* __restrict__ lbatch,
                                 float* __restrict__ out) {
  __shared__ float Bs[2][64 * BSTR];
  const int mBlock = blockIdx.x * 128;
  const int pBase = blockIdx.y * 64;
  v8f acc[4] = {};
  block_gemm(x, W1w, mBlock, pBase, Bs, acc);

  const int lane = threadIdx.x & 31;
  const int wave = threadIdx.x >> 5;
  const int rhi = (lane >> 4) * 8;
  const int n = lane & 15;
#pragma unroll
  for (int r = 0; r < 8; ++r) {
    const int row = mBlock + wave * 16 + r + rhi;
    const int b = lbatch[row];
    const int s = b * NL + label[row];
    const unsigned diff = cntB[b] - cntBL[s];
    const float scale = (diff > 0u) ? (1.0f / (float)diff) : 0.0f;
    const float* __restrict__ Yb = Y + (size_t)b * PDIM;
    const float* __restrict__ Ys = Y + (size_t)(NB + s) * PDIM;
    float* __restrict__ orow = out + (size_t)row * PDIM;
#pragma unroll
    for (int t = 0; t < 4; ++t) {
      const int p = pBase + t * 16 + n;
      orow[p] = acc[t][r] + W1b[p] + W2b[p] + scale * (Yb[p] - Ys[p]);
    }
  }
}

// ---------------------------------------------------------------------------
extern "C" void kernel_launch(void* const* d_in, const int* in_sizes, int n_in,
                              void* d_out, int out_size, void* d_ws, size_t ws_size,
                              hipStream_t stream) {
  (void)in_sizes; (void)n_in; (void)out_size; (void)ws_size;
  const float* x     = (const float*)d_in[0];
  const int*   label = (const int*)d_in[1];
  const int*   lb    = (const int*)d_in[2];
  const float* W1w   = (const float*)d_in[3];
  const float* W1b   = (const float*)d_in[4];
  const float* W2w   = (const float*)d_in[5];
  const float* W2b   = (const float*)d_in[6];
  float* out = (float*)d_out;

  // Workspace layout (S padded to SROWS_P rows; pad rows stay zero)
  const size_t S_BYTES   = (size_t)SROWS_P * DDIM * sizeof(float);
  const size_t CNT_BYTES = (size_t)(NB + NSEG) * sizeof(unsigned);
  char* ws = (char*)d_ws;
  float*    S     = (float*)ws;
  unsigned* cntB  = (unsigned*)(ws + S_BYTES);
  unsigned* cntBL = cntB + NB;
  float*    Y     = (float*)(ws + S_BYTES + CNT_BYTES);  // SROWS_P * PDIM

  // Zero accumulation scratch (atomics need a clean start every call).
  (void)hipMemsetAsync(ws, 0, S_BYTES + CNT_BYTES, stream);

  seg_count_kernel<<<NROWS / 256, 256, 0, stream>>>(label, lb, cntB, cntBL);
  seg_scatter_kernel<<<NROWS, 256, 0, stream>>>(x, label, lb, S);
  batch_reduce_kernel<<<(NB * DDIM) / 256, 256, 0, stream>>>(S);

  // Y = S @ W2^T : 26 x 4 blocks of 128 rows x 64 cols
  gemm_seg_kernel<<<dim3(SROWS_P / 128, PDIM / 64), 256, 0, stream>>>(S, W2w, Y);

  // out = x @ W1^T + biases + gathered/scaled correction : 128 x 4 blocks
  gemm_main_kernel<<<dim3(NROWS / 128, PDIM / 64), 256, 0, stream>>>(
      x, W1w, W1b, W2b, Y, cntB, cntBL, label, lb, out);
}